// SubnetInteractionBlock_33732673143451
// MI455X (gfx1250) — compile-verified
//
#include <hip/hip_runtime.h>

#define D 512
#define S 2048

typedef __attribute__((ext_vector_type(2))) float v2f;
typedef __attribute__((ext_vector_type(8))) float v8f;

// ---------------------------------------------------------------- zero ws
__global__ void zero_ws_kernel(float* __restrict__ sums, float* __restrict__ counts) {
    int i = blockIdx.x * blockDim.x + threadIdx.x;
    if (i < S * D) sums[i] = 0.0f;
    if (i < S)     counts[i] = 0.0f;
}

// ------------------------------------------------- scatter-add into sums/counts
// One thread per (row, 4-col chunk): float4 read of x, 4 L2 atomics into the
// 4MB accumulator (L2-resident on 192MB L2).
__global__ void scatter_kernel(const float* __restrict__ x,
                               const int* __restrict__ sid,
                               float* __restrict__ sums,
                               float* __restrict__ counts, int n) {
    long long i = (long long)blockIdx.x * blockDim.x + threadIdx.x;
    long long total = (long long)n * (D / 4);
    if (i >= total) return;
    int row = (int)(i / (D / 4));
    int c4  = (int)(i % (D / 4)) * 4;
    int s   = sid[row];
    const float4 v = *(const float4*)(x + (size_t)row * D + c4);
    float* dst = sums + (size_t)s * D + c4;
    atomicAdd(dst + 0, v.x);
    atomicAdd(dst + 1, v.y);
    atomicAdd(dst + 2, v.z);
    atomicAdd(dst + 3, v.w);
    if (c4 == 0) atomicAdd(counts + s, 1.0f);
}

// ---------------------------------------------------------- sums -> mean (in place)
__global__ void mean_kernel(float* __restrict__ sums, const float* __restrict__ counts) {
    int i = blockIdx.x * blockDim.x + threadIdx.x;
    if (i >= S * D) return;
    float cnt = counts[i / D];
    sums[i] = sums[i] / fmaxf(cnt, 1.0f);
}

// ------------------------------------------------- f32 WMMA GEMM: C = op(A@B + bias)
// A: [S, D] row-major, B: [D, D] row-major, C: [S, D]. One wave owns a 16x16
// output tile; K stepped by 4 using V_WMMA_F32_16X16X4_F32 (f32 in, f32 acc).
// Fragment layouts per ISA 7.12.2:
//   A 16x4:  lanes 0-15 -> M=lane, VGPR j = K=j;     lanes 16-31 -> K=2+j
//   B 4x16:  lanes 0-15 -> N=lane, VGPR j = K=j row; lanes 16-31 -> K=2+j row
//   C 16x16: VGPR r = row (m0 + r + 8*(lane>=16)), col = n0 + lane%16
__global__ __launch_bounds__(128)
void gemm_wmma_kernel(const float* __restrict__ A, const float* __restrict__ B,
                      const float* __restrict__ bias, float* __restrict__ C,
                      int relu) {
    const int lane = threadIdx.x & 31;
    const int wave = threadIdx.x >> 5;
    const int m0 = (blockIdx.y * 4 + wave) * 16;
    const int n0 = blockIdx.x * 16;
    const int lg = lane >> 4;   // 0: lanes 0-15, 1: lanes 16-31
    const int lm = lane & 15;

    v8f c = {};
    for (int k = 0; k < D; k += 4) {
        const int ka = k + 2 * lg;
        v2f a = *(const v2f*)(A + (size_t)(m0 + lm) * D + ka);
        v2f b;
        b.x = B[(size_t)(ka + 0) * D + n0 + lm];
        b.y = B[(size_t)(ka + 1) * D + n0 + lm];
        c = __builtin_amdgcn_wmma_f32_16x16x4_f32(
                /*neg_a=*/false, a, /*neg_b=*/false, b,
                /*c_mod=*/(short)0, c, /*reuse_a=*/false, /*reuse_b=*/false);
    }

    const int col = n0 + lm;
    const float bv = bias[col];
#pragma unroll
    for (int r = 0; r < 8; ++r) {
        const int row = m0 + r + 8 * lg;
        float v = c[r] + bv;
        if (relu) v = fmaxf(v, 0.0f);
        C[(size_t)row * D + col] = v;
    }
}

// ---------------------- fused gather + residual + LayerNorm (one block per row)
__global__ __launch_bounds__(256)
void ln_kernel(const float* __restrict__ x, const int* __restrict__ sid,
               const float* __restrict__ f2, const float* __restrict__ gamma,
               const float* __restrict__ beta, float* __restrict__ out, int n) {
    const int row = blockIdx.x;
    if (row >= n) return;
    const int t = threadIdx.x;
    const int s = sid[row];

    const float v0 = x[(size_t)row * D + t]       + f2[(size_t)s * D + t];
    const float v1 = x[(size_t)row * D + t + 256] + f2[(size_t)s * D + t + 256];

    __shared__ float s_sum[256];
    __shared__ float s_sq[256];
    s_sum[t] = v0 + v1;
    s_sq[t]  = v0 * v0 + v1 * v1;
    __syncthreads();
#pragma unroll
    for (int off = 128; off > 0; off >>= 1) {
        if (t < off) {
            s_sum[t] += s_sum[t + off];
            s_sq[t]  += s_sq[t + off];
        }
        __syncthreads();
    }
    __shared__ float s_mu, s_rstd;
    if (t == 0) {
        const float mu  = s_sum[0] * (1.0f / D);
        const float var = s_sq[0] * (1.0f / D) - mu * mu;
        s_mu   = mu;
        s_rstd = rsqrtf(var + 1e-5f);
    }
    __syncthreads();
    const float mu = s_mu, rs = s_rstd;
    out[(size_t)row * D + t]       = (v0 - mu) * rs * gamma[t]       + beta[t];
    out[(size_t)row * D + t + 256] = (v1 - mu) * rs * gamma[t + 256] + beta[t + 256];
}

// ---------------------------------------------------------------- launcher
extern "C" void kernel_launch(void* const* d_in, const int* in_sizes, int n_in,
                              void* d_out, int out_size, void* d_ws, size_t ws_size,
                              hipStream_t stream) {
    const float* x     = (const float*)d_in[0];
    const int*   sid   = (const int*)d_in[1];
    const float* W1    = (const float*)d_in[2];
    const float* b1    = (const float*)d_in[3];
    const float* W2    = (const float*)d_in[4];
    const float* b2    = (const float*)d_in[5];
    const float* gamma = (const float*)d_in[6];
    const float* beta  = (const float*)d_in[7];
    float* out = (float*)d_out;
    const int n = in_sizes[1];   // number of rows (100000)

    // workspace layout: [sums/feat/f2 : S*D][h : S*D][counts : S]
    float* sums   = (float*)d_ws;
    float* h      = sums + (size_t)S * D;
    float* counts = h + (size_t)S * D;

    zero_ws_kernel<<<(S * D + 255) / 256, 256, 0, stream>>>(sums, counts);

    long long scatter_items = (long long)n * (D / 4);
    int scatter_blocks = (int)((scatter_items + 255) / 256);
    scatter_kernel<<<scatter_blocks, 256, 0, stream>>>(x, sid, sums, counts, n);

    mean_kernel<<<(S * D + 255) / 256, 256, 0, stream>>>(sums, counts);

    dim3 ggrid(D / 16, S / 64);                 // 32 x 32 blocks, 4 waves/block
    gemm_wmma_kernel<<<ggrid, 128, 0, stream>>>(sums, W1, b1, h, /*relu=*/1);
    gemm_wmma_kernel<<<ggrid, 128, 0, stream>>>(h, W2, b2, sums, /*relu=*/0);

    ln_kernel<<<n, 256, 0, stream>>>(x, sid, sums, gamma, beta, out, n);
}